// IntegrationLayer_29489245454921
// MI455X (gfx1250) — compile-verified
//
#include <hip/hip_runtime.h>
#include <hip/hip_bf16.h>

// ---- problem constants (match reference) ----
#define H_DIM   768
#define NH      12
#define DH      64
#define SCALE   0.125f   // 1/sqrt(64)

typedef __attribute__((ext_vector_type(2))) float v2f;
typedef __attribute__((ext_vector_type(8))) float v8f;

// ---------------------------------------------------------------------------
// Kernel 1: zero the softmax denominator accumulator
// ---------------------------------------------------------------------------
__global__ void zero_f32(float* __restrict__ p, int n) {
    int i = blockIdx.x * blockDim.x + threadIdx.x;
    if (i < n) p[i] = 0.0f;
}

// ---------------------------------------------------------------------------
// Kernel 2: O = X @ W^T + b  for q/k/v (blockIdx.z selects which).
// One wave computes a 32x32 f32 output tile via V_WMMA_F32_16X16X4_F32:
// 2 A-fragments x 2 B-fragments -> 4 WMMAs per K=4 step, reusing each
// loaded fragment twice (8 FLOP/byte from L2 instead of 4).
//
// A (16x4 f32) lane layout: lane L<16 -> row M=L, VGPRs = {K0,K1};
//                           lane L>=16 -> row M=L-16, VGPRs = {K2,K3}.
// B (4x16) mirrors with columns across lanes. D: VGPR i -> row i (lanes 0-15)
// and row i+8 (lanes 16-31), column = lane&15.
// ---------------------------------------------------------------------------
__global__ void qkv_wmma_gemm(const float* __restrict__ hs,
                              const float* __restrict__ Wq, const float* __restrict__ bq,
                              const float* __restrict__ Wk, const float* __restrict__ bk,
                              const float* __restrict__ Wv, const float* __restrict__ bv,
                              float* __restrict__ q, float* __restrict__ k,
                              float* __restrict__ v, int nrows) {
    const float* W;  const float* bias;  float* O;
    if (blockIdx.z == 0)      { W = Wq; bias = bq; O = q; }
    else if (blockIdx.z == 1) { W = Wk; bias = bk; O = k; }
    else                      { W = Wv; bias = bv; O = v; }

    const int tilesN = H_DIM / 32;                 // 24 tiles of 32 along output dim
    int wave  = blockIdx.x * (blockDim.x >> 5) + (threadIdx.x >> 5);
    int tileM = wave / tilesN;                     // 32-row band
    int tileN = wave % tilesN;                     // 32-col band
    if (tileM * 32 >= nrows) return;

    int lane = threadIdx.x & 31;
    int half = lane >> 4;          // 0: lanes 0-15, 1: lanes 16-31
    int l15  = lane & 15;
    int kOff = half * 2;

    // A rows for the two 16-row fragments, B rows (= output cols, W is [O,K])
    const float* __restrict__ A0 = hs + (size_t)(tileM * 32 + l15)      * H_DIM + kOff;
    const float* __restrict__ A1 = A0 + (size_t)16 * H_DIM;
    const float* __restrict__ B0 = W  + (size_t)(tileN * 32 + l15)      * H_DIM + kOff;
    const float* __restrict__ B1 = B0 + (size_t)16 * H_DIM;

    v8f acc00 = {}, acc01 = {}, acc10 = {}, acc11 = {};
    #pragma unroll 4
    for (int kk = 0; kk < H_DIM; kk += 4) {
        v2f a0 = *(const v2f*)(A0 + kk);   // 8B-aligned global_load_b64
        v2f a1 = *(const v2f*)(A1 + kk);
        v2f b0 = *(const v2f*)(B0 + kk);
        v2f b1 = *(const v2f*)(B1 + kk);
        acc00 = __builtin_amdgcn_wmma_f32_16x16x4_f32(false, a0, false, b0,
                                                      (short)0, acc00, false, false);
        acc01 = __builtin_amdgcn_wmma_f32_16x16x4_f32(false, a0, false, b1,
                                                      (short)0, acc01, false, false);
        acc10 = __builtin_amdgcn_wmma_f32_16x16x4_f32(false, a1, false, b0,
                                                      (short)0, acc10, false, false);
        acc11 = __builtin_amdgcn_wmma_f32_16x16x4_f32(false, a1, false, b1,
                                                      (short)0, acc11, false, false);
    }

    float bb0 = bias[tileN * 32 + l15];
    float bb1 = bias[tileN * 32 + 16 + l15];
    int   row0 = tileM * 32 + half * 8;            // rows for acc0x
    int   row1 = row0 + 16;                        // rows for acc1x
    float* O00 = O + (size_t)row0 * H_DIM + tileN * 32 + l15;
    float* O01 = O00 + 16;
    float* O10 = O + (size_t)row1 * H_DIM + tileN * 32 + l15;
    float* O11 = O10 + 16;
    #pragma unroll
    for (int i = 0; i < 8; ++i) {
        O00[(size_t)i * H_DIM] = acc00[i] + bb0;
        O01[(size_t)i * H_DIM] = acc01[i] + bb1;
        O10[(size_t)i * H_DIM] = acc10[i] + bb0;
        O11[(size_t)i * H_DIM] = acc11[i] + bb1;
    }
}

// ---------------------------------------------------------------------------
// Kernel 3: per-edge exp-scores + scatter-add denominator.
// One wave per edge; DH=64 split as lane & lane+32; 12 heads sequential.
// q/k live in L2 (6.3 MB each), so gathers are cheap.
// ---------------------------------------------------------------------------
__global__ void edge_scores(const float* __restrict__ q, const float* __restrict__ k,
                            const float* __restrict__ rel_key,
                            const int* __restrict__ esrc, const int* __restrict__ etgt,
                            const int* __restrict__ epos,
                            float* __restrict__ scores, float* __restrict__ denom,
                            int nedges) {
    int wave = blockIdx.x * (blockDim.x >> 5) + (threadIdx.x >> 5);
    int lane = threadIdx.x & 31;
    if (wave >= nedges) return;

    int s = esrc[wave];
    int t = etgt[wave];
    int p = epos[wave];

    const float* __restrict__ qb = q + (size_t)t * H_DIM;
    const float* __restrict__ kb = k + (size_t)s * H_DIM;
    float r0 = rel_key[p * DH + lane];
    float r1 = rel_key[p * DH + lane + 32];

    #pragma unroll
    for (int h = 0; h < NH; ++h) {
        int base = h * DH;
        float partial = qb[base + lane]      * (kb[base + lane]      + r0)
                      + qb[base + lane + 32] * (kb[base + lane + 32] + r1);
        // wave32 butterfly reduction
        #pragma unroll
        for (int off = 16; off > 0; off >>= 1)
            partial += __shfl_xor(partial, off, 32);
        if (lane == 0) {
            float sc = expf(partial * SCALE);
            scores[(size_t)wave * NH + h] = sc;
            atomicAdd(&denom[(size_t)t * NH + h], sc);
        }
    }
}

// ---------------------------------------------------------------------------
// Kernel 4: normalize scores in place -> attention weights.
// (scores is recomputed from scratch every call, so this is deterministic.)
// ---------------------------------------------------------------------------
__global__ void edge_normalize(float* __restrict__ scores, const float* __restrict__ denom,
                               const int* __restrict__ etgt, int nedges) {
    int i = blockIdx.x * blockDim.x + threadIdx.x;
    if (i >= nedges * NH) return;
    int e = i / NH;
    int h = i - e * NH;
    scores[i] = scores[i] / denom[(size_t)etgt[e] * NH + h];
}

// ---------------------------------------------------------------------------
// Kernel 5: out[e,h,d] = (v[src,h,d] + rel_val[pos,d]) * attn[e,h]
// This is the HBM-bound phase (402 MB store) -> float4 streaming stores.
// ---------------------------------------------------------------------------
__global__ void edge_weighted(const float* __restrict__ v, const float* __restrict__ rel_val,
                              const float* __restrict__ attn,
                              const int* __restrict__ esrc, const int* __restrict__ epos,
                              float* __restrict__ out, long total4) {
    long idx = (long)blockIdx.x * blockDim.x + threadIdx.x;
    if (idx >= total4) return;
    long base = idx * 4;                 // element index into [E, NH, DH]
    int e = (int)(base / (NH * DH));
    int r = (int)(base - (long)e * (NH * DH));
    int h = r >> 6;
    int d = r & (DH - 1);

    float a = attn[(size_t)e * NH + h];
    const float4 vv = *(const float4*)(v + (size_t)esrc[e] * H_DIM + r);
    const float4 rv = *(const float4*)(rel_val + epos[e] * DH + d);
    float4 o;
    o.x = (vv.x + rv.x) * a;
    o.y = (vv.y + rv.y) * a;
    o.z = (vv.z + rv.z) * a;
    o.w = (vv.w + rv.w) * a;
    *(float4*)(out + base) = o;
}

// ---------------------------------------------------------------------------
extern "C" void kernel_launch(void* const* d_in, const int* in_sizes, int n_in,
                              void* d_out, int out_size, void* d_ws, size_t ws_size,
                              hipStream_t stream) {
    const float* hs      = (const float*)d_in[0];
    const float* Wq      = (const float*)d_in[1];
    const float* bq      = (const float*)d_in[2];
    const float* Wk      = (const float*)d_in[3];
    const float* bk      = (const float*)d_in[4];
    const float* Wv      = (const float*)d_in[5];
    const float* bv      = (const float*)d_in[6];
    const float* rel_key = (const float*)d_in[7];
    const float* rel_val = (const float*)d_in[8];
    const int*   esrc    = (const int*)d_in[9];
    const int*   etgt    = (const int*)d_in[10];
    // d_in[11] = edges_type (unused by the reference)
    const int*   epos    = (const int*)d_in[12];

    const int N = in_sizes[0] / H_DIM;     // 2048 flattened nodes
    const int E = in_sizes[9];             // 131072 edges

    // workspace layout (floats)
    float* ws     = (float*)d_ws;
    float* q      = ws;                                  // N*768
    float* k      = q + (size_t)N * H_DIM;               // N*768
    float* v      = k + (size_t)N * H_DIM;               // N*768
    float* scores = v + (size_t)N * H_DIM;               // E*12
    float* denom  = scores + (size_t)E * NH;             // N*12

    // 1) zero denom
    {
        int n = N * NH;
        zero_f32<<<(n + 255) / 256, 256, 0, stream>>>(denom, n);
    }

    // 2) q/k/v GEMMs via f32 WMMA: 8 waves per block, one 32x32 tile per wave
    {
        int tiles = (N / 32) * (H_DIM / 32);             // 64 * 24 = 1536
        dim3 grid((tiles + 7) / 8, 1, 3);
        qkv_wmma_gemm<<<grid, 256, 0, stream>>>(hs, Wq, bq, Wk, bk, Wv, bv,
                                                q, k, v, N);
    }

    // 3) edge scores + denominator scatter-add (one wave per edge)
    {
        int blocks = (E + 7) / 8;                        // 8 waves per 256-thread block
        edge_scores<<<blocks, 256, 0, stream>>>(q, k, rel_key, esrc, etgt, epos,
                                                scores, denom, E);
    }

    // 4) normalize scores -> attention
    {
        int n = E * NH;
        edge_normalize<<<(n + 255) / 256, 256, 0, stream>>>(scores, denom, etgt, E);
    }

    // 5) weighted values (streaming 402 MB store)
    {
        long total4 = (long)E * NH * DH / 4;             // 25,165,824 float4s
        int blocks = (int)((total4 + 255) / 256);
        edge_weighted<<<blocks, 256, 0, stream>>>(v, rel_val, scores, esrc, epos,
                                                  (float*)d_out, total4);
    }
}